// GATGuard_53953379172810
// MI455X (gfx1250) — compile-verified
//
#include <hip/hip_runtime.h>
#include <math.h>

#define SIM_THRESH 0.1f
#define NEG_SLOPE  0.01f

typedef __attribute__((ext_vector_type(2))) float v2f;
typedef __attribute__((ext_vector_type(8))) float v8f;

// ---------------------------------------------------------------------------
// 1) Per-node inverse L2 norm: inv_n[i] = 1 / max(||x_i||, 1e-12)
//    One wave per node, float4 loads, shfl-xor tree reduce (wave32).
// ---------------------------------------------------------------------------
__global__ __launch_bounds__(256) void node_inv_norm_kernel(
    const float* __restrict__ X, float* __restrict__ inv_n, int N, int D) {
  int node = blockIdx.x * (blockDim.x >> 5) + (threadIdx.x >> 5);
  if (node >= N) return;
  int lane = threadIdx.x & 31;
  const float4* x4 = (const float4*)(X + (size_t)node * D);
  int D4 = D >> 2;
  float acc = 0.f;
  for (int d = lane; d < D4; d += 32) {
    float4 v = x4[d];
    acc += v.x * v.x + v.y * v.y + v.z * v.z + v.w * v.w;
  }
#pragma unroll
  for (int off = 16; off; off >>= 1) acc += __shfl_xor(acc, off);
  if (lane == 0) inv_n[node] = 1.0f / fmaxf(sqrtf(acc), 1e-12f);
}

// ---------------------------------------------------------------------------
// 2) Zero the per-row accumulators (rowsum, deg)
// ---------------------------------------------------------------------------
__global__ __launch_bounds__(256) void zero2_kernel(float* __restrict__ a,
                                                    float* __restrict__ b, int N) {
  int i = blockIdx.x * blockDim.x + threadIdx.x;
  if (i < N) { a[i] = 0.f; b[i] = 0.f; }
}

// ---------------------------------------------------------------------------
// 3) Per-edge thresholded cosine similarity + row accumulations.
//    One wave per edge; gathers are L2-resident (X fits in 192MB L2).
// ---------------------------------------------------------------------------
__global__ __launch_bounds__(256) void edge_sim_kernel(
    const float* __restrict__ X, const float* __restrict__ inv_n,
    const int* __restrict__ row, const int* __restrict__ col,
    float* __restrict__ sim, float* __restrict__ rowsum,
    float* __restrict__ deg, int E, int D) {
  int e = blockIdx.x * (blockDim.x >> 5) + (threadIdx.x >> 5);
  if (e >= E) return;
  int lane = threadIdx.x & 31;
  int r = row[e], c = col[e];
  const float4* xr = (const float4*)(X + (size_t)r * D);
  const float4* xc = (const float4*)(X + (size_t)c * D);
  int D4 = D >> 2;
  float acc = 0.f;
  for (int d = lane; d < D4; d += 32) {
    float4 a = xr[d], b = xc[d];
    acc += a.x * b.x + a.y * b.y + a.z * b.z + a.w * b.w;
  }
#pragma unroll
  for (int off = 16; off; off >>= 1) acc += __shfl_xor(acc, off);
  float s = acc * inv_n[r] * inv_n[c];
  if (s < SIM_THRESH) s = 0.f;   // threshold (negatives and small sims -> 0)
  if (lane == 0) {
    sim[e] = s;
    if (s != 0.f) {
      atomicAdd(rowsum + r, s);   // thresholded sims are >= 0.1, so |s| == s
      atomicAdd(deg + r, 1.0f);   // att != 0 exactly when s != 0
    }
  }
}

// ---------------------------------------------------------------------------
// 4) Projection z = x @ W via f32 WMMA (V_WMMA_F32_16X16X4_F32).
//    Compile-time Din/HID/Dout so every inner-loop load is an
//    immediate-offset global_load off a hoisted base pointer (no per-iter
//    64-bit address math). One wave per 16x16 output tile.
//    ISA lane layouts (wave32):
//      A 16x4 : lanes 0-15 -> M=lane, (K=0,1); lanes 16-31 -> M=lane-16, (K=2,3)
//      B 4x16 : VGPR0 = row K=0 (lanes 0-15) / K=2 (lanes 16-31); VGPR1 = K=1/K=3
//      C/D    : vgpr r -> rows r (lanes 0-15) and r+8 (lanes 16-31), N=lane&15
// ---------------------------------------------------------------------------
template <int Din, int HID, int Dout>
__global__ __launch_bounds__(256) void gemm_wmma_f32_kernel(
    const float* __restrict__ X,   // [N, Din] row-major
    const float* __restrict__ W,   // [Dout/HID, Din, HID] row-major
    float* __restrict__ Z,         // [N, Dout] row-major
    int tilesM) {
  constexpr int tilesN = Dout / 16;
  int wave  = threadIdx.x >> 5;
  int tid   = blockIdx.x * (blockDim.x >> 5) + wave;
  int total = tilesM * tilesN;
  if (tid >= total) return;            // wave-uniform: EXEC stays all-ones
  int tileM = tid % tilesM;
  int tileN = tid / tilesM;
  int lane  = threadIdx.x & 31;
  int half  = lane >> 4;               // K-half selector (0 -> K0/1, 1 -> K2/3)
  int l16   = lane & 15;

  int rowm = tileM * 16 + l16;         // A row handled by this lane
  int ncol = tileN * 16 + l16;         // global output column for this lane
  int head = ncol / HID;
  int o    = ncol - head * HID;

  // Hoist the lane-dependent K-half into the base pointers once; all loads
  // below then use compile-time byte offsets.
  const float* xr = X + (size_t)rowm * Din + 2 * half;
  const float* wc = W + ((size_t)head * Din + 2 * half) * HID + o;

  v8f c = {};
#pragma unroll 8
  for (int k = 0; k < Din; k += 4) {
    v2f a, b;
    a.x = xr[k];                       // global_load_b64 xr + 4k (imm offset)
    a.y = xr[k + 1];
    b.x = wc[(size_t)k * HID];         // global_load_b32, imm offset 4*k*HID
    b.y = wc[(size_t)(k + 1) * HID];
    c = __builtin_amdgcn_wmma_f32_16x16x4_f32(
        /*neg_a=*/false, a, /*neg_b=*/false, b,
        /*c_mod=*/(short)0, c, /*reuse_a=*/false, /*reuse_b=*/false);
  }

  float* op = Z + (size_t)(tileM * 16 + half * 8) * Dout + tileN * 16 + l16;
#pragma unroll
  for (int r = 0; r < 8; ++r) op[(size_t)r * Dout] = c[r];
}

// ---------------------------------------------------------------------------
// 5) Self-loop term initializes the accumulator: out = exp(1/(deg+1)) * z
//    DoutShift = log2(Dout) -> node index by shift, no integer division.
// ---------------------------------------------------------------------------
template <int DoutShift>
__global__ __launch_bounds__(256) void self_init_kernel(
    const float* __restrict__ z, const float* __restrict__ deg,
    float* __restrict__ out, int n) {
  int i = blockIdx.x * blockDim.x + threadIdx.x;
  if (i >= n) return;
  int node = i >> DoutShift;
  float wself = expf(1.0f / (deg[node] + 1.0f));
  out[i] = wself * z[i];
}

// ---------------------------------------------------------------------------
// 6) Edge scatter: out[row] += exp(sim/rowsum[row]) * z[col]  (segment_sum)
//    One wave per edge; f32 global atomics (L2-resident destination).
// ---------------------------------------------------------------------------
__global__ __launch_bounds__(256) void edge_scatter_kernel(
    const float* __restrict__ z, const float* __restrict__ sim,
    const float* __restrict__ rowsum,
    const int* __restrict__ row, const int* __restrict__ col,
    float* __restrict__ out, int E, int Dout) {
  int e = blockIdx.x * (blockDim.x >> 5) + (threadIdx.x >> 5);
  if (e >= E) return;
  float s = sim[e];
  if (s == 0.0f) return;               // dropped nnz contributes nothing
  int r = row[e], c = col[e];
  float w = expf(s / rowsum[r]);       // rowsum[r] >= s > 0 guaranteed here
  int lane = threadIdx.x & 31;
  const float* zc = z + (size_t)c * Dout;
  float* orow = out + (size_t)r * Dout;
  for (int d = lane; d < Dout; d += 32) atomicAdd(orow + d, w * zc[d]);
}

// ---------------------------------------------------------------------------
// 7) LeakyReLU in place
// ---------------------------------------------------------------------------
__global__ __launch_bounds__(256) void leaky_relu_kernel(float* __restrict__ x, int n) {
  int i = blockIdx.x * blockDim.x + threadIdx.x;
  if (i < n) {
    float v = x[i];
    x[i] = v > 0.f ? v : NEG_SLOPE * v;
  }
}

// ---------------------------------------------------------------------------
// Host-side layer driver (stream-ordered kernel chain)
// ---------------------------------------------------------------------------
template <int Din, int H, int HID, int DoutShift>
static void run_layer(const float* x, const float* W, float* z, float* out,
                      const int* row, const int* col,
                      float* inv_n, float* rowsum, float* deg, float* sim,
                      int N, int E, bool act, hipStream_t stream) {
  constexpr int Dout = H * HID;
  static_assert((1 << DoutShift) == Dout, "DoutShift must be log2(Dout)");
  node_inv_norm_kernel<<<dim3((N + 7) / 8), dim3(256), 0, stream>>>(x, inv_n, N, Din);
  zero2_kernel<<<dim3((N + 255) / 256), dim3(256), 0, stream>>>(rowsum, deg, N);
  edge_sim_kernel<<<dim3((E + 7) / 8), dim3(256), 0, stream>>>(
      x, inv_n, row, col, sim, rowsum, deg, E, Din);
  {
    int tilesM = N / 16;
    int total = tilesM * (Dout / 16);
    gemm_wmma_f32_kernel<Din, HID, Dout><<<dim3((total + 7) / 8), dim3(256), 0, stream>>>(
        x, W, z, tilesM);
  }
  {
    int n = N * Dout;
    self_init_kernel<DoutShift><<<dim3((n + 255) / 256), dim3(256), 0, stream>>>(
        z, deg, out, n);
  }
  edge_scatter_kernel<<<dim3((E + 7) / 8), dim3(256), 0, stream>>>(
      z, sim, rowsum, row, col, out, E, Dout);
  if (act) {
    int n = N * Dout;
    leaky_relu_kernel<<<dim3((n + 255) / 256), dim3(256), 0, stream>>>(out, n);
  }
}

extern "C" void kernel_launch(void* const* d_in, const int* in_sizes, int n_in,
                              void* d_out, int out_size, void* d_ws, size_t ws_size,
                              hipStream_t stream) {
  (void)n_in; (void)out_size; (void)ws_size;
  const float* x  = (const float*)d_in[0];   // [N, 128]
  const float* W0 = (const float*)d_in[1];   // [4, 128, 64]
  const float* W1 = (const float*)d_in[2];   // [4, 256, 64]
  const float* W2 = (const float*)d_in[3];   // [1, 256, 16]
  const int*  row = (const int*)d_in[4];     // [E]
  const int*  col = (const int*)d_in[5];     // [E]

  const int IN = 128, HID = 64, H = 4;
  const int N = in_sizes[0] / IN;
  const int E = in_sizes[4];
  const int DMID = H * HID;                  // 256

  // Carve workspace (all buffers fully rewritten each call).
  float* ws = (float*)d_ws;
  float* h0     = ws; ws += (size_t)N * DMID;  // layer0 output  [N,256]
  float* h1     = ws; ws += (size_t)N * DMID;  // layer1 output  [N,256]
  float* z      = ws; ws += (size_t)N * DMID;  // projection scratch
  float* inv_n  = ws; ws += N;
  float* rowsum = ws; ws += N;
  float* deg    = ws; ws += N;
  float* sim    = ws; ws += E;

  // Layer 0: x[N,128] -> h0[N,256], leaky-relu
  run_layer<128, 4, 64, 8>(x,  W0, z, h0, row, col, inv_n, rowsum, deg, sim,
                           N, E, true,  stream);
  // Layer 1: h0[N,256] -> h1[N,256], leaky-relu
  run_layer<256, 4, 64, 8>(h0, W1, z, h1, row, col, inv_n, rowsum, deg, sim,
                           N, E, true,  stream);
  // Layer 2: h1[N,256] -> d_out[N,16], no activation
  run_layer<256, 1, 16, 4>(h1, W2, z, (float*)d_out, row, col,
                           inv_n, rowsum, deg, sim, N, E, false, stream);
}